// DPGC_13357348290767
// MI455X (gfx1250) — compile-verified
//
#include <hip/hip_runtime.h>
#include <hip/hip_bf16.h>

typedef __attribute__((ext_vector_type(2))) float v2f;
typedef __attribute__((ext_vector_type(8))) float v8f;

#define N_NODES 20000
#define N_EDGES 200000
#define BN_SC 0.9999950000374997f   // 1/sqrt(1+1e-5)

// ---------------------------------------------------------------------------
// Generic fp32 WMMA GEMM:  C[M,Nout] = A[M,K] @ B (+bias) (+relu*BN)
//  BTRANS=true : B is W[Nout,K] row-major (ldb = row stride) -> C = A @ W^T
//  BTRANS=false: B is [K,Nout] row-major (ldb)               -> C = A @ B
//  One wave computes a 32x64 strip (2 M-tiles x 4 N-tiles = 8 WMMA tiles),
//  K stepped by 4 with register double-buffered fragments so the loadcnt
//  wait lands after the 8 wmmas of each step.
//  M must be a multiple of 32 (true for all call sites: 20000, 200000).
// ---------------------------------------------------------------------------
template<int OPFLAG, bool BTRANS>
__global__ void gemm_wmma(const float* __restrict__ A, int lda,
                          const float* __restrict__ B, int ldb,
                          const float* __restrict__ bias,
                          float* __restrict__ C, int ldc,
                          int M, int Nout, int K)
{
  const int lane = threadIdx.x & 31;
  const int wid  = blockIdx.x * (blockDim.x >> 5) + (threadIdx.x >> 5);
  const int ntm  = M >> 5;               // 32 rows per wave
  const int ntn  = (Nout + 63) >> 6;
  if (wid >= ntm * ntn) return;          // wave-uniform: EXEC stays all-ones
  const int tm = wid / ntn, tn = wid % ntn;

  const int kb   = (lane >> 4) << 1;     // 0 (lanes 0-15) or 2 (lanes 16-31)
  const int colL = lane & 15;

  const float* Ap[2];
#pragma unroll
  for (int m = 0; m < 2; ++m)
    Ap[m] = A + (size_t)(tm * 32 + m * 16 + (lane & 15)) * lda + kb;

  const float* Bp[4];
#pragma unroll
  for (int j = 0; j < 4; ++j) {
    int n = tn * 64 + j * 16 + colL;
    int ncj = (n < Nout) ? n : (Nout - 1);   // clamp: keeps loads in-bounds
    if (BTRANS) Bp[j] = B + (size_t)ncj * ldb + kb;
    else        Bp[j] = B + ncj;             // index with (k+kb)*ldb
  }

  v8f acc[2][4];
#pragma unroll
  for (int m = 0; m < 2; ++m)
#pragma unroll
    for (int j = 0; j < 4; ++j)
#pragma unroll
      for (int i = 0; i < 8; ++i) acc[m][j][i] = 0.0f;

  v2f a_cur[2], b_cur[4], a_nxt[2], b_nxt[4];
#pragma unroll
  for (int m = 0; m < 2; ++m) a_cur[m] = *(const v2f*)(Ap[m]);
#pragma unroll
  for (int j = 0; j < 4; ++j) {
    if (BTRANS) b_cur[j] = *(const v2f*)(Bp[j]);
    else { b_cur[j].x = Bp[j][(size_t)kb * ldb]; b_cur[j].y = Bp[j][(size_t)(kb + 1) * ldb]; }
  }

  for (int k = 4; k < K; k += 4) {
    // prefetch next fragments (no dependency on the wmmas below)
#pragma unroll
    for (int m = 0; m < 2; ++m) a_nxt[m] = *(const v2f*)(Ap[m] + k);
#pragma unroll
    for (int j = 0; j < 4; ++j) {
      if (BTRANS) b_nxt[j] = *(const v2f*)(Bp[j] + k);
      else {
        b_nxt[j].x = Bp[j][(size_t)(k + kb) * ldb];
        b_nxt[j].y = Bp[j][(size_t)(k + kb + 1) * ldb];
      }
    }
#pragma unroll
    for (int m = 0; m < 2; ++m)
#pragma unroll
      for (int j = 0; j < 4; ++j)
        acc[m][j] = __builtin_amdgcn_wmma_f32_16x16x4_f32(
            false, a_cur[m], false, b_cur[j], (short)0, acc[m][j], false, false);
#pragma unroll
    for (int m = 0; m < 2; ++m) a_cur[m] = a_nxt[m];
#pragma unroll
    for (int j = 0; j < 4; ++j) b_cur[j] = b_nxt[j];
  }
#pragma unroll
  for (int m = 0; m < 2; ++m)
#pragma unroll
    for (int j = 0; j < 4; ++j)
      acc[m][j] = __builtin_amdgcn_wmma_f32_16x16x4_f32(
          false, a_cur[m], false, b_cur[j], (short)0, acc[m][j], false, false);

#pragma unroll
  for (int m = 0; m < 2; ++m) {
    const int rbase = tm * 32 + m * 16 + ((lane >> 4) << 3);
#pragma unroll
    for (int j = 0; j < 4; ++j) {
      int n = tn * 64 + j * 16 + colL;
      if (n >= Nout) continue;
      float bb = bias ? bias[n] : 0.0f;
#pragma unroll
      for (int i = 0; i < 8; ++i) {
        float v = acc[m][j][i] + bb;
        if (OPFLAG == 1) v = fmaxf(v, 0.0f) * BN_SC;
        C[(size_t)(rbase + i) * ldc + n] = v;
      }
    }
  }
}

// ---------------------------------------------------------------------------
__global__ void fill_f32(float* __restrict__ p, long n, float v)
{
  long i = (long)blockIdx.x * blockDim.x + threadIdx.x;
  if (i < n) p[i] = v;
}

// parser stage 1: out[e,j] = relu(x[e, off:off+3] . p1_w[j] + p1_b[j]) * BN
__global__ void parser1_relu_bn(const float* __restrict__ x, int xoff,
                                const float* __restrict__ w,
                                const float* __restrict__ b,
                                float* __restrict__ out)
{
  long idx = (long)blockIdx.x * blockDim.x + threadIdx.x;
  if (idx >= (long)N_EDGES * 128) return;
  int  j = (int)(idx & 127);
  long e = idx >> 7;
  const float* xe = x + e * 6 + xoff;
  float s = b[j] + xe[0] * w[j * 3 + 0] + xe[1] * w[j * 3 + 1] + xe[2] * w[j * 3 + 2];
  out[idx] = fmaxf(s, 0.0f) * BN_SC;
}

// per-node |es|^2 (es lives in eies columns 512..1023), one wave per node
__global__ void node_sqnorm(const float* __restrict__ eies, float* __restrict__ esn)
{
  int wid  = blockIdx.x * (blockDim.x >> 5) + (threadIdx.x >> 5);
  int lane = threadIdx.x & 31;
  if (wid >= N_NODES) return;
  const float* row = eies + (size_t)wid * 1024 + 512;
  float s = 0.0f;
  for (int j = lane; j < 512; j += 32) { float v = row[j]; s += v * v; }
  for (int o = 16; o > 0; o >>= 1) s += __shfl_xor(s, o, 32);
  if (lane == 0) esn[wid] = s;
}

// edge weight = (cos(h1,h2)+1)/2 without materializing h1/h2; one wave per edge
__global__ void edge_weight(const int* __restrict__ eidx,
                            const float* __restrict__ eies,
                            const float* __restrict__ p2a,
                            const float* __restrict__ p2b,
                            const float* __restrict__ esn,
                            float* __restrict__ w, float* __restrict__ deg)
{
  int wid  = blockIdx.x * (blockDim.x >> 5) + (threadIdx.x >> 5);
  int lane = threadIdx.x & 31;
  if (wid >= N_EDGES) return;
  int s = eidx[wid], t = eidx[N_EDGES + wid];
  const float* pa = p2a + (size_t)wid * 128;
  const float* pb = p2b + (size_t)wid * 128;
  float dot = 0.0f, na = 0.0f, nb = 0.0f;
  for (int j = lane; j < 128; j += 32) {
    float a = pa[j], b = pb[j];
    dot += a * b; na += a * a; nb += b * b;
  }
  const float* esr = eies + (size_t)s * 1024 + 512;
  const float* est = eies + (size_t)t * 1024 + 512;
  for (int j = lane; j < 512; j += 32) dot += esr[j] * est[j];
  for (int o = 16; o > 0; o >>= 1) {
    dot += __shfl_xor(dot, o, 32);
    na  += __shfl_xor(na,  o, 32);
    nb  += __shfl_xor(nb,  o, 32);
  }
  if (lane == 0) {
    float n1 = fmaxf(sqrtf(na + esn[s]), 1e-8f);
    float n2 = fmaxf(sqrtf(nb + esn[t]), 1e-8f);
    float ww = (dot / (n1 * n2) + 1.0f) * 0.5f;
    w[wid] = ww;
    atomicAdd(&deg[s], ww);
  }
}

__global__ void dis_k(const float* __restrict__ deg, float* __restrict__ dis)
{
  int i = blockIdx.x * blockDim.x + threadIdx.x;
  if (i < N_NODES) { float d = deg[i]; dis[i] = (d > 0.0f) ? rsqrtf(d) : 0.0f; }
}

__global__ void norm_k(const int* __restrict__ eidx, const float* __restrict__ w,
                       const float* __restrict__ dis, float* __restrict__ nrm)
{
  int e = blockIdx.x * blockDim.x + threadIdx.x;
  if (e < N_EDGES) nrm[e] = dis[eidx[e]] * w[e] * dis[eidx[N_EDGES + e]];
}

// prop: out[tgt, f] += -norm[e] * z[src, f]   (out must be pre-zeroed, F=16)
__global__ void prop_k(const int* __restrict__ eidx, const float* __restrict__ nrm,
                       const float* __restrict__ z, int ldz, int zoff,
                       float* __restrict__ out)
{
  long idx = (long)blockIdx.x * blockDim.x + threadIdx.x;
  if (idx >= (long)N_EDGES * 16) return;
  int  f = (int)(idx & 15);
  long e = idx >> 4;
  int s = eidx[e], t = eidx[N_EDGES + e];
  atomicAdd(out + (size_t)t * 16 + f, -nrm[e] * z[(size_t)s * ldz + zoff + f]);
}

// h = relu(B0 - B2 + T1 + 2*T2), stored into jk column block `layer`
__global__ void cheb_combine(const float* __restrict__ B,
                             const float* __restrict__ T1,
                             const float* __restrict__ T2,
                             float* __restrict__ jk, int layer)
{
  long idx = (long)blockIdx.x * blockDim.x + threadIdx.x;
  if (idx >= (long)N_NODES * 16) return;
  int  f = (int)(idx & 15);
  long n = idx >> 4;
  float v = B[n * 48 + f] - B[n * 48 + 32 + f] + T1[idx] + 2.0f * T2[idx];
  jk[n * 64 + layer * 16 + f] = fmaxf(v, 0.0f);
}

__global__ void copy_es(const float* __restrict__ eies, float* __restrict__ out)
{
  long idx = (long)blockIdx.x * blockDim.x + threadIdx.x;
  if (idx >= (long)N_NODES * 512) return;
  long n = idx >> 9;
  int  f = (int)(idx & 511);
  out[idx] = eies[n * 1024 + 512 + f];
}

// ---------------------------------------------------------------------------
static inline dim3 grid_for(long threads) { return dim3((unsigned)((threads + 255) / 256)); }

static inline dim3 gemm_grid(int M, int Nout)
{
  long waves = (long)(M >> 5) * ((Nout + 63) >> 6);
  return dim3((unsigned)((waves * 32 + 255) / 256));
}

extern "C" void kernel_launch(void* const* d_in, const int* in_sizes, int n_in,
                              void* d_out, int out_size, void* d_ws, size_t ws_size,
                              hipStream_t stream)
{
  (void)in_sizes; (void)n_in; (void)out_size; (void)ws_size;
  const int   N = N_NODES, E = N_EDGES;
  const float* imgf  = (const float*)d_in[0];
  const int*   eidx  = (const int*)  d_in[1];
  const float* nonim = (const float*)d_in[2];
  const float* EI_w = (const float*)d_in[3],  *EI_b = (const float*)d_in[4];
  const float* ES_w = (const float*)d_in[5],  *ES_b = (const float*)d_in[6];
  const float* DE_w = (const float*)d_in[7],  *DE_b = (const float*)d_in[8];
  const float* chw[4] = { (const float*)d_in[9], (const float*)d_in[10],
                          (const float*)d_in[11], (const float*)d_in[12] };
  const float* l1_w = (const float*)d_in[13], *l1_b = (const float*)d_in[14];
  const float* l2_w = (const float*)d_in[15], *l2_b = (const float*)d_in[16];
  const float* s1_w = (const float*)d_in[17], *s1_b = (const float*)d_in[18];
  const float* s2_w = (const float*)d_in[19], *s2_b = (const float*)d_in[20];
  const float* p1_w = (const float*)d_in[21], *p1_b = (const float*)d_in[22];
  const float* p2_w = (const float*)d_in[23], *p2_b = (const float*)d_in[24];

  float* outp = (float*)d_out;
  float* o_label = outp;                       // [N,2]
  float* o_site  = outp + (size_t)N * 2;       // [N,20]
  float* o_es    = outp + (size_t)N * 22;      // [N,512]
  float* o_rec   = outp + (size_t)N * 534;     // [N,512]

  // ---- workspace layout (floats, 256-float aligned chunks) ----
  float* wsf = (float*)d_ws;
  size_t off = 0;
  auto take = [&](size_t nElem) { size_t o = off; off += (nElem + 255) & ~(size_t)255; return o; };
  float* eies = wsf + take((size_t)N * 1024);  // [N, ei|es]
  float* p1b  = wsf + take((size_t)E * 128);   // parser1 temp (reused, aliased by clf)
  float* p2a  = wsf + take((size_t)E * 128);
  float* p2b  = wsf + take((size_t)E * 128);
  float* esn  = wsf + take(N);
  float* wgt  = wsf + take(E);
  float* deg  = wsf + take(N);
  float* dis  = wsf + take(N);
  float* nrm  = wsf + take(E);
  float* Bb   = wsf + take((size_t)N * 48);    // [N, B0|B1|B2]
  float* T1   = wsf + take((size_t)N * 16);
  float* T2a  = wsf + take((size_t)N * 16);
  float* T2b  = wsf + take((size_t)N * 16);
  float* jk   = wsf + take((size_t)N * 64);
  float* clfb = p1b;                           // [N,256]; p1b is dead by then

  // ---- ei / es : [N,2000] @ [512,2000]^T, into concat buffer ----
  gemm_wmma<0, true><<<gemm_grid(N, 512), 256, 0, stream>>>(
      imgf, 2000, EI_w, 2000, EI_b, eies,       1024, N, 512, 2000);
  gemm_wmma<0, true><<<gemm_grid(N, 512), 256, 0, stream>>>(
      imgf, 2000, ES_w, 2000, ES_b, eies + 512, 1024, N, 512, 2000);

  // ---- edge MLP (two halves): p1 then p2 GEMM ----
  parser1_relu_bn<<<grid_for((long)E * 128), 256, 0, stream>>>(nonim, 0, p1_w, p1_b, p1b);
  gemm_wmma<0, true><<<gemm_grid(E, 128), 256, 0, stream>>>(
      p1b, 128, p2_w, 128, p2_b, p2a, 128, E, 128, 128);
  parser1_relu_bn<<<grid_for((long)E * 128), 256, 0, stream>>>(nonim, 3, p1_w, p1_b, p1b);
  gemm_wmma<0, true><<<gemm_grid(E, 128), 256, 0, stream>>>(
      p1b, 128, p2_w, 128, p2_b, p2b, 128, E, 128, 128);

  // ---- edge weights, degree, normalization ----
  node_sqnorm<<<grid_for((long)N * 32), 256, 0, stream>>>(eies, esn);
  fill_f32<<<grid_for(N), 256, 0, stream>>>(deg, N, 0.0f);
  edge_weight<<<grid_for((long)E * 32), 256, 0, stream>>>(eidx, eies, p2a, p2b, esn, wgt, deg);
  dis_k<<<grid_for(N), 256, 0, stream>>>(deg, dis);
  norm_k<<<grid_for(E), 256, 0, stream>>>(eidx, wgt, dis, nrm);

  // ---- reconstruct = [ei|es] @ DE_w^T + b ----
  gemm_wmma<0, true><<<gemm_grid(N, 512), 256, 0, stream>>>(
      eies, 1024, DE_w, 1024, DE_b, o_rec, 512, N, 512, 1024);

  // ---- ChebConv x4 (project-then-propagate; prop commutes with feature matmul)
  //      out = B0 - B2 + prop(B1) + 2*prop(prop(B2)),  B_k = x @ W[k]
  for (int l = 0; l < 4; ++l) {
    const float* Ain = (l == 0) ? eies : (jk + (size_t)(l - 1) * 16);
    int lda = (l == 0) ? 1024 : 64;
    int Kin = (l == 0) ? 512 : 16;
    for (int k = 0; k < 3; ++k) {
      gemm_wmma<0, false><<<gemm_grid(N, 16), 256, 0, stream>>>(
          Ain, lda, chw[l] + (size_t)k * Kin * 16, 16, nullptr,
          Bb + k * 16, 48, N, 16, Kin);
    }
    fill_f32<<<grid_for((long)N * 16), 256, 0, stream>>>(T1, (long)N * 16, 0.0f);
    prop_k<<<grid_for((long)E * 16), 256, 0, stream>>>(eidx, nrm, Bb, 48, 16, T1);
    fill_f32<<<grid_for((long)N * 16), 256, 0, stream>>>(T2a, (long)N * 16, 0.0f);
    prop_k<<<grid_for((long)E * 16), 256, 0, stream>>>(eidx, nrm, Bb, 48, 32, T2a);
    fill_f32<<<grid_for((long)N * 16), 256, 0, stream>>>(T2b, (long)N * 16, 0.0f);
    prop_k<<<grid_for((long)E * 16), 256, 0, stream>>>(eidx, nrm, T2a, 16, 0, T2b);
    cheb_combine<<<grid_for((long)N * 16), 256, 0, stream>>>(Bb, T1, T2b, jk, l);
  }

  // ---- label classifier: jk[N,64] -> 256 (relu*BN) -> 2 ----
  gemm_wmma<1, true><<<gemm_grid(N, 256), 256, 0, stream>>>(
      jk, 64, l1_w, 64, l1_b, clfb, 256, N, 256, 64);
  gemm_wmma<0, true><<<gemm_grid(N, 2), 256, 0, stream>>>(
      clfb, 256, l2_w, 256, l2_b, o_label, 2, N, 2, 256);

  // ---- site classifier: ei[N,512] -> 256 (relu*BN) -> 20 ----
  gemm_wmma<1, true><<<gemm_grid(N, 256), 256, 0, stream>>>(
      eies, 1024, s1_w, 512, s1_b, clfb, 256, N, 256, 512);
  gemm_wmma<0, true><<<gemm_grid(N, 20), 256, 0, stream>>>(
      clfb, 256, s2_w, 256, s2_b, o_site, 20, N, 20, 256);

  // ---- es output copy (strided) ----
  copy_es<<<grid_for((long)N * 512), 256, 0, stream>>>(eies, o_es);
}